// Quantizer_20753281974729
// MI455X (gfx1250) — compile-verified
//
#include <hip/hip_runtime.h>

// Problem constants (from the reference): B=4, H=16, L=4096, D=64, S=512.
#define BDIM 4
#define HDIM 16
#define LDIM 4096
#define DDIM 64
#define SDIM 512

typedef __attribute__((ext_vector_type(2))) float v2f;
typedef __attribute__((ext_vector_type(4))) float v4f;
typedef __attribute__((ext_vector_type(8))) float v8f;

static constexpr size_t OH_ELEMS = (size_t)BDIM * HDIM * LDIM * SDIM; // 134,217,728

// ---------------------------------------------------------------------------
// Kernel 1: codebook normalization + |c|^2 reduction.
// One wave32 per (h,s) row: 32 lanes x float2 covers D=64.
// Writes c into d_out tail (it's an output) and c2 into workspace.
// ---------------------------------------------------------------------------
__global__ __launch_bounds__(256) void vq_codebook_kernel(
    const float* __restrict__ c_sum,   // [H,S,D]
    const float* __restrict__ c_count, // [H,S]
    float* __restrict__ c_out,         // [H,S,D]
    float* __restrict__ c2_out)        // [H,S]
{
    int wave = (blockIdx.x * blockDim.x + threadIdx.x) >> 5;
    int lane = threadIdx.x & 31;
    if (wave >= HDIM * SDIM) return;
    int hs = wave;

    float cnt = fmaxf(c_count[hs], 1e-6f);   // jnp.clip(count, 1e-6)
    v2f v = *(const v2f*)(c_sum + (size_t)hs * DDIM + lane * 2);
    v2f c;
    c.x = v.x / cnt;
    c.y = v.y / cnt;
    *(v2f*)(c_out + (size_t)hs * DDIM + lane * 2) = c;

    float sq = c.x * c.x + c.y * c.y;
    #pragma unroll
    for (int m = 16; m >= 1; m >>= 1) sq += __shfl_xor(sq, m, 32);
    if (lane == 0) c2_out[hs] = sq;
}

// ---------------------------------------------------------------------------
// Kernel 2: WMMA inner products + running argmin + one-hot write.
// One wave per 16-row l-tile of one (b,h). A tile (16x64 f32) preloaded as
// 16 K=4 fragments; sweep 32 s-tiles with V_WMMA_F32_16X16X4_F32 chains.
// argmin uses score = c2[s] - 2*xc[l,s]  (|x|^2 row-constant dropped).
// ---------------------------------------------------------------------------
__global__ __launch_bounds__(256) void vq_argmin_kernel(
    const float* __restrict__ x,   // [B*H, L, D]
    const float* __restrict__ c,   // [H, S, D]
    const float* __restrict__ c2,  // [H, S]
    float* __restrict__ onehot)    // [B*H, L, S]
{
    constexpr int LT = LDIM / 16; // 256 l-tiles per (b,h)
    int wave  = (blockIdx.x * blockDim.x + threadIdx.x) >> 5;
    int lane  = threadIdx.x & 31;
    int ltile = wave % LT;
    int bh    = wave / LT;
    int h     = bh % HDIM;
    int l0    = ltile * 16;

    int half = lane >> 4;     // which 16-lane half
    int q    = lane & 15;     // M for A; N for B/C/D
    int kb   = half * 2;      // K offset within each K=4 step (ISA A/B layout)

    // Preload full A tile: a[t] holds A[q][4t+kb .. 4t+kb+1] (x is row-major [l][d])
    const float* px = x + ((size_t)bh * LDIM + l0 + q) * DDIM + kb;
    v2f a[16];
    #pragma unroll
    for (int t = 0; t < 16; ++t) a[t] = *(const v2f*)(px + 4 * t);

    float best[8];
    int   bidx[8];
    #pragma unroll
    for (int r = 0; r < 8; ++r) { best[r] = 3.4e38f; bidx[r] = 0; }

    const float* cb  = c  + (size_t)h * SDIM * DDIM;
    const float* c2b = c2 + h * SDIM;

    for (int s0 = 0; s0 < SDIM; s0 += 16) {
        // B fragments: B[k][n] = c[s0+n][k]  (c row-major [s][d] -> float2 loads)
        const float* pc = cb + (size_t)(s0 + q) * DDIM + kb;
        v2f bf[16];
        #pragma unroll
        for (int t = 0; t < 16; ++t) bf[t] = *(const v2f*)(pc + 4 * t);

        v8f acc = {};
        #pragma unroll
        for (int t = 0; t < 16; ++t) {
            // D = A x B + C ; 8 args: (neg_a, A, neg_b, B, c_mod, C, reuse_a, reuse_b)
            acc = __builtin_amdgcn_wmma_f32_16x16x4_f32(
                false, a[t], false, bf[t], (short)0, acc, false, false);
        }

        float c2v = c2b[s0 + q]; // column constant (N = q)
        #pragma unroll
        for (int r = 0; r < 8; ++r) {
            float score = fmaf(-2.0f, acc[r], c2v);
            if (score < best[r]) { best[r] = score; bidx[r] = s0 + q; }
        }
    }

    // Cross-lane argmin within each 16-lane half; ties -> lower index
    // (matches jnp.argmin first-min semantics).
    #pragma unroll
    for (int r = 0; r < 8; ++r) {
        #pragma unroll
        for (int m = 8; m >= 1; m >>= 1) {
            float ov = __shfl_xor(best[r], m, 32);
            int   oi = __shfl_xor(bidx[r], m, 32);
            if (ov < best[r] || (ov == best[r] && oi < bidx[r])) {
                best[r] = ov; bidx[r] = oi;
            }
        }
    }

    // One-hot write: half owns rows l0 + 8*half + r (C/D layout: M = r + 8*half).
    // Lane q writes float4 at s = j*64 + q*4 -> 16 consecutive lanes store
    // 256 contiguous bytes per instruction; full 512-float row in 8 stores.
    float* outb = onehot + ((size_t)bh * LDIM + l0 + 8 * half) * SDIM;
    #pragma unroll
    for (int r = 0; r < 8; ++r) {
        float* rowp = outb + (size_t)r * SDIM;
        int target = bidx[r];
        #pragma unroll
        for (int j = 0; j < 8; ++j) {
            int sbase = j * 64 + q * 4;
            v4f v;
            v.x = (target == sbase + 0) ? 1.0f : 0.0f;
            v.y = (target == sbase + 1) ? 1.0f : 0.0f;
            v.z = (target == sbase + 2) ? 1.0f : 0.0f;
            v.w = (target == sbase + 3) ? 1.0f : 0.0f;
            *(v4f*)(rowp + sbase) = v;
        }
    }
}

extern "C" void kernel_launch(void* const* d_in, const int* in_sizes, int n_in,
                              void* d_out, int out_size, void* d_ws, size_t ws_size,
                              hipStream_t stream) {
    const float* x       = (const float*)d_in[0]; // [B,H,L,D]
    const float* c_sum   = (const float*)d_in[1]; // [H,S,D]
    const float* c_count = (const float*)d_in[2]; // [H,S]

    float* onehot = (float*)d_out;            // output 0: [B,H,L,S]
    float* c_out  = onehot + OH_ELEMS;        // output 1: [H,S,D]
    float* c2     = (float*)d_ws;             // scratch:  [H,S] = 32 KB

    // Kernel 1: 8192 waves (one per codebook row), 8 waves/block.
    vq_codebook_kernel<<<(HDIM * SDIM) / 8, 256, 0, stream>>>(c_sum, c_count, c_out, c2);

    // Kernel 2: B*H*(L/16) = 16384 waves, 8 waves/block -> 2048 blocks.
    int nwaves = BDIM * HDIM * (LDIM / 16);
    vq_argmin_kernel<<<nwaves / 8, 256, 0, stream>>>(x, c_out, c2, onehot);
}